// ImageLC2_27986006900769
// MI455X (gfx1250) — compile-verified
//
#include <hip/hip_runtime.h>
#include <hip/hip_bf16.h>

typedef float v2f __attribute__((ext_vector_type(2)));
typedef float v8f __attribute__((ext_vector_type(8)));

#define DIM     153
#define PSZ     51
#define CPAD    20           // (51 - 11) / 2
#define CW      11           // 2*RADIUS + 1
#define COUNT   1331         // 11^3
#define KPAD    1344         // COUNT rounded up to 4*NWAVES (42 * 32)
#define RSTRIDE 1348         // KPAD + 4 dwords: row starts 4 banks apart
#define NPATCH  108          // 4 * 3^3
#define NWAVES  8
#define NTHREADS (32 * NWAVES)
#define KITER   (KPAD / (4 * NWAVES))   // 42: identical for every wave
#define ALPHA   0.001f
#define BETA    0.01f

// One block = 8 waves, 4 patches.
// Phase 1: stage M = [m; g; 1; b] for 4 patches into LDS rows (row = 4p+c),
//          K-padded with zeros so the WMMA loop needs no tail handling.
// Phase 2: each wave accumulates a partial 16x16 Gram M*M^T over a strided
//          K-slice with v_wmma_f32_16x16x4_f32; constant trip count + one
//          base register + immediate DS offsets -> fully unrolled, loads
//          pipelined under the WMMA accumulation chain.
// Phase 3: sum partial Grams, lanes 0..3 solve the 3x3 regularized normal
//          equations per patch and emit the clipped symmetry score.
__global__ __launch_bounds__(NTHREADS)
void lc2_gram_kernel(const float* __restrict__ us,
                     const float* __restrict__ mr,
                     float* __restrict__ s_out) {
    __shared__ __align__(16) float V[16][RSTRIDE];  // operand rows (~86 KB)
    __shared__ float G[NWAVES][16][16];             // partial Grams (8 KB)

    const int tid   = (int)threadIdx.x;
    const int pbase = blockIdx.x * 4;

    // ---------------- Phase 1: stage operand matrix ----------------
#pragma unroll
    for (int p = 0; p < 4; ++p) {
        const int P  = pbase + p;
        const int b  = P / 27;
        const int pi = P % 27;
        const int z0 = (pi / 9) * PSZ;
        const int y0 = ((pi / 3) % 3) * PSZ;
        const int x0 = (pi % 3) * PSZ;
        const size_t bas = (((size_t)(b * DIM + z0) * DIM + y0) * DIM + x0);
        const float* __restrict__ mrb = mr + bas;
        const float* __restrict__ usb = us + bas;

        for (int k = tid; k < KPAD; k += NTHREADS) {
            float m = 0.0f, g = 0.0f, one = 0.0f, bv = 0.0f;
            if (k < COUNT) {
                const int d   = k / (CW * CW);
                const int h   = (k / CW) % CW;
                const int w   = k % CW;
                const int off = ((CPAD + d) * DIM + (CPAD + h)) * DIM
                              + (CPAD + w);
                // central differences: neighbors of the central 11^3 region
                // never leave the 51^3 patch, so the reference's per-patch
                // zero padding is irrelevant here. All six neighbor loads
                // fold into immediate offsets of one base address.
                m  = mrb[off];
                bv = usb[off];
                const float gx = mrb[off - 1]         - mrb[off + 1];
                const float gy = mrb[off - DIM]       - mrb[off + DIM];
                const float gz = mrb[off - DIM * DIM] - mrb[off + DIM * DIM];
                g   = __builtin_sqrtf(gx * gx + gy * gy + gz * gz);
                one = 1.0f;   // zero in the K-tail keeps sum(1) == COUNT
            }
            V[4 * p + 0][k] = m;
            V[4 * p + 1][k] = g;
            V[4 * p + 2][k] = one;
            V[4 * p + 3][k] = bv;
        }
    }
    __syncthreads();

    // ---------------- Phase 2: WMMA Gram accumulation ----------------
    const int lane = tid & 31;
    const int wv   = tid >> 5;
    const int half = lane >> 4;    // K sub-pair select (A & B layouts)
    const int row  = lane & 15;    // matrix row (A) == column (B) = 4p+c
    // One base register per lane; every iteration is base + immediate.
    const float* __restrict__ frag0 = &V[row][wv * 4 + 2 * half];

    v8f acc = {};
#pragma unroll
    for (int i = 0; i < KITER; ++i) {
        // lane fragment = M[row, ka], M[row, ka+1]; 8-byte aligned,
        // DS immediate offset = 128*i bytes (fits the 16-bit offset field)
        const v2f ab = *(const v2f*)(frag0 + 32 * i);
        acc = __builtin_amdgcn_wmma_f32_16x16x4_f32(
            /*neg_a=*/false, ab, /*neg_b=*/false, ab,
            /*c_mod=*/(short)0, acc, /*reuse_a=*/false, /*reuse_b=*/false);
    }

    // C/D layout: acc[v] is (row = v + 8*half, col = lane&15)
#pragma unroll
    for (int v = 0; v < 8; ++v)
        G[wv][v + 8 * half][row] = acc[v];
    __syncthreads();

    // ---------------- Phase 3: per-patch 3x3 solve ----------------
    if (wv == 0 && lane < 4) {
        const int o = lane * 4;
        const float inv = 1.0f / (float)COUNT;

        float g00 = 0.f, g01 = 0.f, g02 = 0.f, g03 = 0.f;
        float g11 = 0.f, g12 = 0.f, g13 = 0.f;
        float g22 = 0.f, g23 = 0.f, g33 = 0.f;
#pragma unroll
        for (int wvi = 0; wvi < NWAVES; ++wvi) {
            g00 += G[wvi][o + 0][o + 0];
            g01 += G[wvi][o + 0][o + 1];
            g02 += G[wvi][o + 0][o + 2];
            g03 += G[wvi][o + 0][o + 3];
            g11 += G[wvi][o + 1][o + 1];
            g12 += G[wvi][o + 1][o + 2];
            g13 += G[wvi][o + 1][o + 3];
            g22 += G[wvi][o + 2][o + 2];
            g23 += G[wvi][o + 2][o + 3];
            g33 += G[wvi][o + 3][o + 3];
        }

        // C = Gram/count + alpha*I (3x3, symmetric); Atb = cross/count
        const float c00 = g00 * inv + ALPHA;
        const float c01 = g01 * inv;
        const float c02 = g02 * inv;
        const float c11 = g11 * inv + ALPHA;
        const float c12 = g12 * inv;
        const float c22 = g22 * inv + ALPHA;   // g22 == COUNT exactly
        const float a0  = g03 * inv;
        const float a1  = g13 * inv;
        const float a2  = g23 * inv;           // mean(b)
        const float bb  = g33 * inv;           // mean(b^2)

        // coeff = C^-1 * Atb via adjugate (symmetric 3x3)
        const float adj00 = c11 * c22 - c12 * c12;
        const float adj01 = c02 * c12 - c01 * c22;
        const float adj02 = c01 * c12 - c02 * c11;
        const float adj11 = c00 * c22 - c02 * c02;
        const float adj12 = c01 * c02 - c00 * c12;
        const float adj22 = c00 * c11 - c01 * c01;
        const float det   = c00 * adj00 + c01 * adj01 + c02 * adj02;
        const float rdet  = 1.0f / det;
        const float x = (adj00 * a0 + adj01 * a1 + adj02 * a2) * rdet;
        const float y = (adj01 * a0 + adj11 * a1 + adj12 * a2) * rdet;
        const float z = (adj02 * a0 + adj12 * a1 + adj22 * a2) * rdet;

        const float quad = c00 * x * x + c11 * y * y + c22 * z * z
                         + 2.0f * (c01 * x * y + c02 * x * z + c12 * y * z);
        const float var  = bb - a2 * a2;
        const float dist = bb + quad - 2.0f * (x * a0 + y * a1 + z * a2);
        const float sym  = (var - dist) / fmaxf(var, BETA);
        s_out[pbase + lane] = fminf(fmaxf(sym, 0.0f), 1.0f);
    }
}

// Deterministic mean of the 108 per-patch scores (no FP atomics).
__global__ __launch_bounds__(32)
void lc2_reduce_kernel(const float* __restrict__ s_in,
                       float* __restrict__ out, int n) {
    __shared__ float sm[32];
    float p = 0.0f;
    for (int i = (int)threadIdx.x; i < n; i += 32) p += s_in[i];
    sm[threadIdx.x] = p;
    __syncthreads();
    if (threadIdx.x == 0) {
        float s = 0.0f;
        for (int i = 0; i < 32; ++i) s += sm[i];
        *out = s / (float)n;
    }
}

extern "C" void kernel_launch(void* const* d_in, const int* in_sizes, int n_in,
                              void* d_out, int out_size, void* d_ws, size_t ws_size,
                              hipStream_t stream) {
    (void)in_sizes; (void)n_in; (void)out_size; (void)ws_size;
    const float* us = (const float*)d_in[0];
    const float* mr = (const float*)d_in[1];
    float* s_ws = (float*)d_ws;   // 108 per-patch scores

    lc2_gram_kernel<<<NPATCH / 4, NTHREADS, 0, stream>>>(us, mr, s_ws);
    lc2_reduce_kernel<<<1, 32, 0, stream>>>(s_ws, (float*)d_out, NPATCH);
}